// QMeasureDMClassif_75161927680260
// MI455X (gfx1250) — compile-verified
//
#include <hip/hip_runtime.h>

// MI455X / gfx1250 implementation.
//
//   rho_y[b,j,k] = sum_{p,m} AT[b,p,m] * rho[p,j,m,k],  AT[b] = X[b]^T @ X[b]^T
//   out[b] = rho_y[b] / trace(rho_y[b])
//
// Kernel 1: batched f32 WMMA GEMM (V_WMMA_F32_16X16X4_F32), 137.4 GFLOP, compute-bound.
//           LDS tiles are K-innermost so every WMMA operand fragment is a single
//           naturally aligned ds_load_b64 (no VGPR repacking movs).
// Kernel 2: per-p LDS-staged contraction, rho read exactly once (~688 MB total -> ~30us floor).
// Kernel 3: trace + normalize.

typedef __attribute__((ext_vector_type(2))) float v2f;
typedef __attribute__((ext_vector_type(8))) float v8f;

#define DXDIM 1024
#define DYDIM 10
#define NBATCH 64

// LDS row stride (in floats) for the K-innermost operand tiles.
// 20 = 16 (K) + 4 pad: 20c mod 64 is distinct for c=0..15 (conflict-free b64
// fragment reads per half-wave), and row*20*4 bytes is 16B-aligned (float4 staging).
#define S1 20

// ---------------------------------------------------------------------------
// Kernel 1: AT[b][p][m] = sum_i X[b][i][p] * X[b][m][i]   (i.e. (X@X)^T per b)
// GEMM form: C[r=p][c=m] = sum_k Aop[r][k]*Bop[k][c],
//            Aop[r][k] = X[k][r],  Bop[k][c] = X[c][k].
// Workgroup: 256 thr = 8 waves (4x2), tile 128x128; wave tile 32x64 (2x4 WMMA tiles).
// ---------------------------------------------------------------------------
__global__ __launch_bounds__(256)
void k1_batched_gemm_t(const float* __restrict__ X, float* __restrict__ AT) {
  const int b  = blockIdx.z;
  const int P0 = blockIdx.y * 128;   // output row block (p)
  const int M0 = blockIdx.x * 128;   // output col block (m)
  const float* Xb  = X  + (size_t)b * DXDIM * DXDIM;
  float*       ATb = AT + (size_t)b * DXDIM * DXDIM;

  // K-innermost operand tiles: frag (kk, kk+1) is one contiguous float2.
  __shared__ float As2[128 * S1];  // As2[r][kk] = Aop[r][kk] = X[k0+kk][P0+r]
  __shared__ float Bs2[128 * S1];  // Bs2[c][kk] = Bop[kk][c] = X[M0+c][k0+kk]

  const int tid   = threadIdx.x;
  const int lane  = tid & 31;
  const int wave  = tid >> 5;       // 0..7
  const int waveM = wave >> 1;      // 0..3 -> +32*waveM rows
  const int waveN = wave & 1;       // 0..1 -> +64*waveN cols
  const int half  = lane >> 4;      // 0/1 (K-pair select per WMMA f32 A/B layout)
  const int l16   = lane & 15;

  // Staging coordinates (fixed per thread).
  const int linA = tid * 4;         // A-slab: 16 rows(k) x 128 cols(p)
  const int rowA = linA >> 7;       // 0..7 (and +8)
  const int colA = linA & 127;
  const int fB   = tid * 2;         // B-slab: 128 rows(m) x 16 floats(k)
  const int rowB = fB >> 2;         // 0..127
  const int kpB  = (fB & 3) * 4;    // 0 or 8 (two float4: kpB, kpB+4)

  v8f acc[2][4];
#pragma unroll
  for (int i = 0; i < 2; i++)
#pragma unroll
    for (int j = 0; j < 4; j++) acc[i][j] = {};

  for (int k0 = 0; k0 < DXDIM; k0 += 16) {
    // ---- stage As2: global rows are k (contiguous in p); scatter K-innermost
    {
      float4 v0 = *(const float4*)(Xb + (size_t)(k0 + rowA) * DXDIM + P0 + colA);
      float4 v1 = *(const float4*)(Xb + (size_t)(k0 + rowA + 8) * DXDIM + P0 + colA);
      const float* a0 = (const float*)&v0;
      const float* a1 = (const float*)&v1;
#pragma unroll
      for (int i = 0; i < 4; i++) {
        As2[(colA + i) * S1 + rowA]     = a0[i];
        As2[(colA + i) * S1 + rowA + 8] = a1[i];
      }
    }
    // ---- stage Bs2: global rows are m (contiguous in k); K-innermost = direct float4
    {
      float4 v0 = *(const float4*)(Xb + (size_t)(M0 + rowB) * DXDIM + k0 + kpB);
      float4 v1 = *(const float4*)(Xb + (size_t)(M0 + rowB) * DXDIM + k0 + kpB + 4);
      *(float4*)&Bs2[rowB * S1 + kpB]     = v0;
      *(float4*)&Bs2[rowB * S1 + kpB + 4] = v1;
    }
    // ---- prefetch next K-slab while this slab computes (global_prefetch_b8)
    if (k0 + 16 < DXDIM) {
      __builtin_prefetch(Xb + (size_t)(k0 + 16 + rowA) * DXDIM + P0 + colA, 0, 3);
      __builtin_prefetch(Xb + (size_t)(k0 + 24 + rowA) * DXDIM + P0 + colA, 0, 3);
      __builtin_prefetch(Xb + (size_t)(M0 + rowB) * DXDIM + k0 + 16 + kpB, 0, 3);
    }
    __syncthreads();

#pragma unroll
    for (int ks = 0; ks < 4; ks++) {
      const int kk = ks * 4 + 2 * half;  // this half-wave's K pair (even -> 8B aligned)
      v2f afrag[2];
#pragma unroll
      for (int tm = 0; tm < 2; tm++) {
        int r = waveM * 32 + tm * 16 + l16;          // A frag: M in lanes, K pair contiguous
        afrag[tm] = *(const v2f*)&As2[r * S1 + kk];  // one ds_load_b64
      }
      v2f bfrag[4];
#pragma unroll
      for (int tn = 0; tn < 4; tn++) {
        int c = waveN * 64 + tn * 16 + l16;          // B frag: N in lanes, K pair contiguous
        bfrag[tn] = *(const v2f*)&Bs2[c * S1 + kk];  // one ds_load_b64
      }
#pragma unroll
      for (int tm = 0; tm < 2; tm++)
#pragma unroll
        for (int tn = 0; tn < 4; tn++)
          acc[tm][tn] = __builtin_amdgcn_wmma_f32_16x16x4_f32(
              /*neg_a=*/false, afrag[tm], /*neg_b=*/false, bfrag[tn],
              /*c_mod=*/(short)0, acc[tm][tn], /*reuse_a=*/false, /*reuse_b=*/false);
    }
    __syncthreads();
  }

  // ---- store: D layout = 8 VGPRs, lane half selects M+8, lanes give N
#pragma unroll
  for (int tm = 0; tm < 2; tm++)
#pragma unroll
    for (int tn = 0; tn < 4; tn++) {
      int colg = M0 + waveN * 64 + tn * 16 + l16;
#pragma unroll
      for (int r = 0; r < 8; r++) {
        int rowg = P0 + waveM * 32 + tm * 16 + r + 8 * half;
        ATb[(size_t)rowg * DXDIM + colg] = acc[tm][tn][r];
      }
    }
}

// ---------------------------------------------------------------------------
// Kernel 2: per-p contraction. One workgroup per p; loops m in slabs of 128.
// Thread t (of 160): owns b-block bg=t/10 (4 batches) x jk-block jg=t%10 (10 jk)
// -> 40 accumulators, 40 FMAs per 14 LDS reads per m.
// rho read exactly once device-wide; A read exactly once device-wide.
// ---------------------------------------------------------------------------
#define S2 132   // LDS row stride (pad vs 64 banks)
__global__ __launch_bounds__(160)
void k2_contract_rho(const float* __restrict__ AT, const float* __restrict__ rho,
                     float* __restrict__ accOut) {
  const int p = blockIdx.x;                       // 0..1023
  __shared__ float Aslab[NBATCH * S2];            // Aslab[b][mm]
  __shared__ float Rslab[DYDIM * DYDIM * S2];     // Rslab[j*10+k][mm]
  const int tid = threadIdx.x;                    // 0..159
  const int bg  = tid / 10;                       // 0..15
  const int jg  = tid % 10;                       // 0..9

  float acc[4][10];
#pragma unroll
  for (int bi = 0; bi < 4; bi++)
#pragma unroll
    for (int ji = 0; ji < 10; ji++) acc[bi][ji] = 0.f;

  const float* rhoP = rho + (size_t)p * (DYDIM * DXDIM * DYDIM); // rho[p,...]

  for (int m0 = 0; m0 < DXDIM; m0 += 128) {
    // stage A: AT[b][p][m0+mm], 128 contiguous floats per b
    for (int idx = tid; idx < NBATCH * 128; idx += 160) {
      int bb = idx >> 7, mm = idx & 127;
      Aslab[bb * S2 + mm] = AT[((size_t)bb * DXDIM + p) * DXDIM + m0 + mm];
    }
    // stage rho: for fixed (p,j), the (m,k) block is 1280 contiguous floats
    for (int f = tid; f < DYDIM * 320; f += 160) {   // float4 granules
      int j = f / 320;
      int o = (f % 320) * 4;
      float4 v = *(const float4*)(rhoP + (size_t)j * (DXDIM * DYDIM) +
                                  (size_t)m0 * DYDIM + o);
      const float* vv = (const float*)&v;
#pragma unroll
      for (int q = 0; q < 4; q++) {
        int oo = o + q;
        int mm = oo / 10, k = oo % 10;
        Rslab[(j * 10 + k) * S2 + mm] = vv[q];
      }
    }
    __syncthreads();

    for (int mm = 0; mm < 128; mm++) {
      float a[4];
#pragma unroll
      for (int bi = 0; bi < 4; bi++) a[bi] = Aslab[(bg * 4 + bi) * S2 + mm];
      float r[10];
#pragma unroll
      for (int ji = 0; ji < 10; ji++) r[ji] = Rslab[(jg * 10 + ji) * S2 + mm];
#pragma unroll
      for (int bi = 0; bi < 4; bi++)
#pragma unroll
        for (int ji = 0; ji < 10; ji++)
          acc[bi][ji] = fmaf(a[bi], r[ji], acc[bi][ji]);
    }
    __syncthreads();
  }

#pragma unroll
  for (int bi = 0; bi < 4; bi++)
#pragma unroll
    for (int ji = 0; ji < 10; ji++)
      atomicAdd(&accOut[(bg * 4 + bi) * 100 + jg * 10 + ji], acc[bi][ji]);
}

// ---------------------------------------------------------------------------
// Kernel 3: trace + normalize.  Kernel 0: zero the atomic accumulator.
// ---------------------------------------------------------------------------
__global__ void k0_zero(float* __restrict__ acc) {
  int i = blockIdx.x * 256 + threadIdx.x;
  if (i < NBATCH * 100) acc[i] = 0.f;
}

__global__ void k3_finalize(const float* __restrict__ acc, float* __restrict__ out) {
  const int b = blockIdx.x;
  __shared__ float tr;
  if (threadIdx.x == 0) {
    float t = 0.f;
#pragma unroll
    for (int j = 0; j < DYDIM; j++) t += acc[b * 100 + j * 10 + j];
    tr = t;
  }
  __syncthreads();
  int jk = threadIdx.x;
  if (jk < 100) out[b * 100 + jk] = acc[b * 100 + jk] / tr;
}

// ---------------------------------------------------------------------------
extern "C" void kernel_launch(void* const* d_in, const int* in_sizes, int n_in,
                              void* d_out, int out_size, void* d_ws, size_t ws_size,
                              hipStream_t stream) {
  const float* X   = (const float*)d_in[0];   // [64,1024,1024] f32
  const float* rho = (const float*)d_in[1];   // [1024,10,1024,10] f32
  float* out = (float*)d_out;                 // [64,10,10] f32

  // ws layout: [0, 256MiB) = AT f32[64][1024][1024]; then f32[6400] accumulator.
  float* AT  = (float*)d_ws;
  float* acc = (float*)((char*)d_ws + (size_t)NBATCH * DXDIM * DXDIM * sizeof(float));

  k0_zero<<<(NBATCH * 100 + 255) / 256, 256, 0, stream>>>(acc);

  dim3 g1(DXDIM / 128, DXDIM / 128, NBATCH);
  k1_batched_gemm_t<<<g1, 256, 0, stream>>>(X, AT);

  k2_contract_rho<<<DXDIM, 160, 0, stream>>>(AT, rho, acc);

  k3_finalize<<<NBATCH, 128, 0, stream>>>(acc, out);
}